// SSM_23742579212440
// MI455X (gfx1250) — compile-verified
//
#include <hip/hip_runtime.h>
#include <hip/hip_bf16.h>
#include <math.h>

// ---------------------------------------------------------------------------
// Mamba-style selective scan block for MI455X (gfx1250, wave32).
//   1) dbc GEMM (fp32 WMMA 16x16x4)  -> LN(48)+SiLU -> split delta16/B/C
//   2) dt  GEMM (fp32 WMMA 16x16x4)  -> +b_dt, LN(256), softplus -> delta
//   3) segmented selective scan (3 passes) to fill the WGPs
// ---------------------------------------------------------------------------

typedef float v2f __attribute__((ext_vector_type(2)));
typedef float v8f __attribute__((ext_vector_type(8)));

#define B_    8
#define L_    2048
#define D_    256
#define RANK_ 16
#define NST_  16
#define ML_   (B_ * L_)           // 16384 rows
#define DBC_  (RANK_ + 2 * NST_)  // 48
#define EPS_  1e-6f
#define SEG_  32
#define TSEG_ (L_ / SEG_)         // 64

// ---------------- FP32 WMMA GEMM: C = A(MxK) * B(KxN), all row-major -------
// One wave per 16x16 output tile. A 16x4 tile: lanes 0-15 hold K={0,1},
// lanes 16-31 hold K={2,3} (ISA 7.12.2). D tile: M = r + 8*half, N = lane%16.
__global__ void wmma_gemm_f32(const float* __restrict__ A,
                              const float* __restrict__ Bw,
                              float* __restrict__ C,
                              int M, int N, int K) {
  const int lane = threadIdx.x & 31;
  const int half = lane >> 4;
  const int lid  = lane & 15;
  const int wid  = (blockIdx.x * blockDim.x + threadIdx.x) >> 5;
  const int tilesN = N >> 4;
  const int tm = wid / tilesN;
  const int tn = wid % tilesN;

  const float* Arow = A + (size_t)(tm * 16 + lid) * K;   // per-lane A row
  const float* Bcol = Bw + tn * 16 + lid;                // per-lane B column
  v8f acc = {};
  for (int k0 = 0; k0 < K; k0 += 4) {
    const int ka = k0 + 2 * half;
    v2f av, bv;
    av.x = Arow[ka];
    av.y = Arow[ka + 1];
    bv.x = Bcol[(size_t)ka * N];
    bv.y = Bcol[(size_t)(ka + 1) * N];
    acc = __builtin_amdgcn_wmma_f32_16x16x4_f32(false, av, false, bv,
                                                (short)0, acc, false, false);
  }
  float* Cp = C + (size_t)(tm * 16 + 8 * half) * N + tn * 16 + lid;
#pragma unroll
  for (int r = 0; r < 8; ++r) Cp[(size_t)r * N] = acc[r];
}

// ---------------- LN(48) + SiLU + split into delta16 / B / C ---------------
__global__ void ln_silu_split(const float* __restrict__ dbc,
                              const float* __restrict__ g1,
                              const float* __restrict__ b1,
                              float* __restrict__ dlt16,
                              float* __restrict__ Bm,
                              float* __restrict__ Cm) {
  const int r = blockIdx.x * blockDim.x + threadIdx.x;
  if (r >= ML_) return;
  const float* x = dbc + (size_t)r * DBC_;
  float m = 0.f;
#pragma unroll
  for (int j = 0; j < DBC_; ++j) m += x[j];
  m *= (1.f / DBC_);
  float v = 0.f;
#pragma unroll
  for (int j = 0; j < DBC_; ++j) { float d = x[j] - m; v += d * d; }
  v *= (1.f / DBC_);
  const float inv = rsqrtf(v + EPS_);
#pragma unroll
  for (int j = 0; j < DBC_; ++j) {
    float y = (x[j] - m) * inv * g1[j] + b1[j];
    float s = y / (1.f + expf(-y));            // SiLU
    if (j < RANK_)            dlt16[(size_t)r * RANK_ + j]          = s;
    else if (j < RANK_ + NST_) Bm[(size_t)r * NST_ + (j - RANK_)]   = s;
    else                       Cm[(size_t)r * NST_ + (j - RANK_ - NST_)] = s;
  }
}

// ---------------- +b_dt, LN(256), softplus -> delta ------------------------
__global__ void ln_softplus(const float* __restrict__ raw,
                            const float* __restrict__ b_dt,
                            const float* __restrict__ g2,
                            const float* __restrict__ b2,
                            float* __restrict__ delta) {
  const int r = blockIdx.x;
  const int j = threadIdx.x;           // 256 threads = 8 waves
  const float x = raw[(size_t)r * D_ + j] + b_dt[j];
  __shared__ float red[D_];
  red[j] = x;
  __syncthreads();
  for (int off = D_ / 2; off > 0; off >>= 1) {
    if (j < off) red[j] += red[j + off];
    __syncthreads();
  }
  const float mean = red[0] * (1.f / D_);
  __syncthreads();
  const float dv = x - mean;
  red[j] = dv * dv;
  __syncthreads();
  for (int off = D_ / 2; off > 0; off >>= 1) {
    if (j < off) red[j] += red[j + off];
    __syncthreads();
  }
  const float var = red[0] * (1.f / D_);
  const float y = dv * rsqrtf(var + EPS_) * g2[j] + b2[j];
  const float sp = (y > 20.f) ? y : log1pf(expf(y));   // softplus
  delta[(size_t)r * D_ + j] = sp;
}

// ---------------- Scan pass 1: per-segment local scan ----------------------
// grid = B_*SEG_, block = 256 (one thread per channel d; 16 states per thread)
__global__ void scan_pass1(const float* __restrict__ delta,
                           const float* __restrict__ u,
                           const float* __restrict__ Bm,
                           const float* __restrict__ A_log,
                           float* __restrict__ h_end,
                           float* __restrict__ Pdec) {
  const int b = blockIdx.x / SEG_;
  const int s = blockIdx.x % SEG_;
  const int d = threadIdx.x;

  float a[NST_], h[NST_], lp[NST_];
#pragma unroll
  for (int n = 0; n < NST_; ++n) {
    a[n] = -expf(A_log[(size_t)d * NST_ + n]);
    h[n] = 0.f;
    lp[n] = 0.f;
  }

  __shared__ float Bs[TSEG_][NST_];
  {
    const float* src = Bm + ((size_t)b * L_ + (size_t)s * TSEG_) * NST_;
    for (int i = threadIdx.x; i < TSEG_ * NST_; i += blockDim.x)
      (&Bs[0][0])[i] = src[i];
  }
  __syncthreads();

  const size_t base = ((size_t)b * L_ + (size_t)s * TSEG_) * D_ + d;
  for (int tt = 0; tt < TSEG_; ++tt) {
    __builtin_prefetch(delta + base + (size_t)(tt + 8) * D_, 0, 0);
    const float dlt = delta[base + (size_t)tt * D_];
    const float uu  = u[base + (size_t)tt * D_];
    const float du  = dlt * uu;
#pragma unroll
    for (int n = 0; n < NST_; ++n) {
      const float xx = dlt * a[n];
      h[n] = expf(xx) * h[n] + du * Bs[tt][n];
      lp[n] += xx;
    }
  }
  const size_t o = (((size_t)b * SEG_ + s) * D_ + d) * NST_;
#pragma unroll
  for (int n = 0; n < NST_; ++n) {
    h_end[o + n] = h[n];
    Pdec[o + n]  = expf(lp[n]);
  }
}

// ---------------- Scan pass 2: sequential segment-prefix fix-up ------------
// grid = B_, block = 256
__global__ void scan_pass2(const float* __restrict__ h_end,
                           const float* __restrict__ Pdec,
                           float* __restrict__ h_in) {
  const int b = blockIdx.x;
  const int d = threadIdx.x;
  float hp[NST_];
#pragma unroll
  for (int n = 0; n < NST_; ++n) hp[n] = 0.f;
  for (int s = 0; s < SEG_; ++s) {
    const size_t o = (((size_t)b * SEG_ + s) * D_ + d) * NST_;
#pragma unroll
    for (int n = 0; n < NST_; ++n) h_in[o + n] = hp[n];
#pragma unroll
    for (int n = 0; n < NST_; ++n) hp[n] = Pdec[o + n] * hp[n] + h_end[o + n];
  }
}

// ---------------- Scan pass 3: replay with true initial states, emit y -----
__global__ void scan_pass3(const float* __restrict__ delta,
                           const float* __restrict__ u,
                           const float* __restrict__ Bm,
                           const float* __restrict__ Cm,
                           const float* __restrict__ A_log,
                           const float* __restrict__ Dvec,
                           const float* __restrict__ h_in,
                           float* __restrict__ out) {
  const int b = blockIdx.x / SEG_;
  const int s = blockIdx.x % SEG_;
  const int d = threadIdx.x;

  float a[NST_], h[NST_];
  const size_t hi = (((size_t)b * SEG_ + s) * D_ + d) * NST_;
#pragma unroll
  for (int n = 0; n < NST_; ++n) {
    a[n] = -expf(A_log[(size_t)d * NST_ + n]);
    h[n] = h_in[hi + n];
  }
  const float Dd = Dvec[d];

  __shared__ float Bs[TSEG_][NST_];
  __shared__ float Cs[TSEG_][NST_];
  {
    const size_t row0 = ((size_t)b * L_ + (size_t)s * TSEG_) * NST_;
    for (int i = threadIdx.x; i < TSEG_ * NST_; i += blockDim.x) {
      (&Bs[0][0])[i] = Bm[row0 + i];
      (&Cs[0][0])[i] = Cm[row0 + i];
    }
  }
  __syncthreads();

  const size_t base = ((size_t)b * L_ + (size_t)s * TSEG_) * D_ + d;
  for (int tt = 0; tt < TSEG_; ++tt) {
    __builtin_prefetch(u + base + (size_t)(tt + 8) * D_, 0, 0);
    const float dlt = delta[base + (size_t)tt * D_];
    const float uu  = u[base + (size_t)tt * D_];
    const float du  = dlt * uu;
    float acc = uu * Dd;
#pragma unroll
    for (int n = 0; n < NST_; ++n) {
      h[n] = expf(dlt * a[n]) * h[n] + du * Bs[tt][n];
      acc += h[n] * Cs[tt][n];
    }
    out[base + (size_t)tt * D_] = acc;
  }
}

// ---------------------------------------------------------------------------
extern "C" void kernel_launch(void* const* d_in, const int* in_sizes, int n_in,
                              void* d_out, int out_size, void* d_ws, size_t ws_size,
                              hipStream_t stream) {
  const float* inputs = (const float*)d_in[0];  // (8,2048,256)
  const float* W_dbc  = (const float*)d_in[1];  // (256,48)
  const float* g1     = (const float*)d_in[2];  // (48)
  const float* b1     = (const float*)d_in[3];  // (48)
  const float* W_dt   = (const float*)d_in[4];  // (16,256)
  const float* b_dt   = (const float*)d_in[5];  // (256)
  const float* g2     = (const float*)d_in[6];  // (256)
  const float* b2     = (const float*)d_in[7];  // (256)
  const float* A_log  = (const float*)d_in[8];  // (256,16)
  const float* Dvec   = (const float*)d_in[9];  // (256)
  float* out = (float*)d_out;                   // (8,2048,256)

  // Workspace layout (floats)
  float* ws       = (float*)d_ws;
  float* dbc_raw  = ws;                               // 16384*48
  float* dlt16    = dbc_raw + (size_t)ML_ * DBC_;     // 16384*16
  float* BmW      = dlt16   + (size_t)ML_ * RANK_;    // 16384*16
  float* CmW      = BmW     + (size_t)ML_ * NST_;     // 16384*16
  float* dt_raw   = CmW     + (size_t)ML_ * NST_;     // 16384*256
  float* deltaW   = dt_raw  + (size_t)ML_ * D_;       // 16384*256
  float* h_end    = deltaW  + (size_t)ML_ * D_;       // 8*32*256*16
  float* PdecW    = h_end   + (size_t)B_ * SEG_ * D_ * NST_;
  float* h_inW    = PdecW   + (size_t)B_ * SEG_ * D_ * NST_;

  // 1) dbc = inputs @ W_dbc : M=16384, N=48, K=256 -> 3072 tiles, 8 waves/blk
  {
    const int tiles = (ML_ / 16) * (DBC_ / 16);       // 3072
    wmma_gemm_f32<<<tiles / 8, 256, 0, stream>>>(inputs, W_dbc, dbc_raw,
                                                 ML_, DBC_, D_);
  }
  // 2) LN(48) + SiLU + split
  ln_silu_split<<<ML_ / 256, 256, 0, stream>>>(dbc_raw, g1, b1,
                                               dlt16, BmW, CmW);
  // 3) dt_raw = delta16 @ W_dt : M=16384, N=256, K=16 -> 16384 tiles
  {
    const int tiles = (ML_ / 16) * (D_ / 16);         // 16384
    wmma_gemm_f32<<<tiles / 8, 256, 0, stream>>>(dlt16, W_dt, dt_raw,
                                                 ML_, D_, RANK_);
  }
  // 4) +b_dt, LN(256), softplus
  ln_softplus<<<ML_, D_, 0, stream>>>(dt_raw, b_dt, g2, b2, deltaW);

  // 5-7) segmented selective scan
  scan_pass1<<<B_ * SEG_, D_, 0, stream>>>(deltaW, inputs, BmW, A_log,
                                           h_end, PdecW);
  scan_pass2<<<B_, D_, 0, stream>>>(h_end, PdecW, h_inW);
  scan_pass3<<<B_ * SEG_, D_, 0, stream>>>(deltaW, inputs, BmW, CmW, A_log,
                                           Dvec, h_inW, out);
}